// CodedNet_23682449670348
// MI455X (gfx1250) — compile-verified
//
#include <hip/hip_runtime.h>
#include <hip/hip_bf16.h>
#include <stdint.h>

// out[b,i,w] = sum_{ch<31} x[b,i,w,ch] * sign(w[((i-ch)&31)*32 + (w&31)])
// Pure bandwidth problem (0.49 FLOP/byte, ~11.5us at 23.3 TB/s): stream the
// 260MB input through LDS with the CDNA5 Tensor Data Mover (one DMA descriptor
// per 256-pixel tile), apply binarized (+-1) coefficients with sign-bit XOR.

constexpr int B_ = 512, H_ = 64, W_ = 64, CH_ = 31;
constexpr int NPIX       = B_ * H_ * W_;          // 2,097,152
constexpr int TILE_PIX   = 256;
constexpr int TILE_BYTES = TILE_PIX * CH_ * 4;    // 31744
constexpr int LDS_BYTES  = 32768;                 // padded for async fallback

typedef unsigned int u32x4 __attribute__((ext_vector_type(4)));
typedef int          i32x8 __attribute__((ext_vector_type(8)));
typedef int          i32x4 __attribute__((ext_vector_type(4)));

__global__ __launch_bounds__(256)
void coded_shift_mask_sum(const float* __restrict__ x,
                          const float* __restrict__ w,
                          float* __restrict__ out)
{
    __shared__ unsigned int colbits[32];
    __shared__ __align__(16) unsigned char tile[LDS_BYTES];

    const int tid = threadIdx.x;

    // Per-(w mod 32) sign bitmask: bit r = signbit(w[r*32 + wc]) -> coeff -1.
    if (tid < 32) {
        unsigned bits = 0u;
        #pragma unroll
        for (int r = 0; r < 32; ++r)
            bits |= (__float_as_uint(w[r * 32 + tid]) >> 31) << r;
        colbits[tid] = bits;
    }

#if defined(__HIP_DEVICE_COMPILE__) && defined(__gfx1250__)
#if __has_builtin(__builtin_amdgcn_tensor_load_to_lds)
    // ---- TDM path: one DMA descriptor fetches the whole 31744B tile ----
    if (tid < 32) {  // wave 0 only: single issue, TENSORcnt is per-wave
        const unsigned long long ga =
            (unsigned long long)(uintptr_t)x +
            (unsigned long long)blockIdx.x * (unsigned)TILE_BYTES;
        const unsigned lds0 = (unsigned)(uintptr_t)&tile[0];
        constexpr unsigned TD0 = (unsigned)(TILE_PIX * CH_);  // 7936 dwords

        u32x4 g0;
        g0[0] = 1u;                                  // count=1, user descriptor
        g0[1] = lds0;                                // lds_addr (bytes)
        g0[2] = (unsigned)(ga & 0xFFFFFFFFull);      // global_addr[31:0]
        g0[3] = (unsigned)(ga >> 32) | (2u << 30);   // global_addr[56:32]|type=2

        i32x8 g1;
        g1[0] = (int)(2u << 16);                     // wg_mask=0, data_size=4B
        g1[1] = (int)((TD0 & 0xFFFFu) << 16);        // tensor_dim0[15:0]
        g1[2] = (int)((TD0 >> 16) | (1u << 16));     // tensor_dim0[31:16], dim1=1
        g1[3] = (int)(TD0 << 16);                    // tile_dim0 = 7936
        g1[4] = 0;                                   // tile_dim1/2 unused
        g1[5] = (int)TD0;                            // tensor_dim0_stride
        g1[6] = 0;
        g1[7] = 0;

        i32x4 g2 = {0, 0, 0, 0};                     // <=2D tensor: groups 2/3
        i32x4 g3 = {0, 0, 0, 0};                     //   disabled

#if __clang_major__ <= 22
        __builtin_amdgcn_tensor_load_to_lds(g0, g1, g2, g3, 0);
#else
        i32x8 g4 = {0, 0, 0, 0, 0, 0, 0, 0};
        __builtin_amdgcn_tensor_load_to_lds(g0, g1, g2, g3, g4, 0);
#endif
        __builtin_amdgcn_s_wait_tensorcnt(0);
    }
#else
    // ---- Fallback: per-lane async global->LDS b128 staging ----
    {
        const unsigned long long xbase = (unsigned long long)(uintptr_t)x;
        const unsigned long long gend  =
            xbase + (unsigned long long)NPIX * CH_ * 4ull - 16ull;
        const unsigned long long tbase =
            xbase + (unsigned long long)blockIdx.x * (unsigned)TILE_BYTES;
        #pragma unroll
        for (int rnd = 0; rnd < 8; ++rnd) {
            unsigned long long ga = tbase + (unsigned)(rnd * 4096 + tid * 16);
            if (ga > gend) ga = gend;  // clamp padded-tail over-read
            unsigned int la =
                (unsigned int)(uintptr_t)&tile[rnd * 4096 + tid * 16];
            asm volatile("global_load_async_to_lds_b128 %0, %1, off"
                         :: "v"(la), "v"(ga) : "memory");
        }
        asm volatile("s_wait_asynccnt 0x0" ::: "memory");
    }
#endif
#else
    // Host-pass / non-gfx1250 semantic fallback (never hot).
    {
        const float* gsrc = x + (size_t)blockIdx.x * TILE_PIX * CH_;
        float* lf = (float*)tile;
        for (int k = tid; k < TILE_PIX * CH_; k += 256) lf[k] = gsrc[k];
    }
#endif
    __syncthreads();

    const int p    = blockIdx.x * TILE_PIX + tid;   // flat (b,i,w) pixel
    const int wc   = p & 31;                        // w mod 32
    const int irow = (p >> 6) & 63;                 // i (row in H)

    // rot bit q == colbits[wc] bit (q + i) mod 32; coeff(ch) lives at
    // bit (i-ch) mod 32 of colbits -> bit (32-ch) mod 32 of rot.
    const unsigned col = colbits[wc];
    const unsigned s   = (unsigned)(irow & 31);
    const unsigned rot = s ? ((col >> s) | (col << (32u - s))) : col;

    const float* pix = (const float*)&tile[tid * (CH_ * 4)];  // stride 31 dw:
    float acc = 0.0f;                                         // bank-conflict-free
    #pragma unroll
    for (int ch = 0; ch < CH_; ++ch) {
        const unsigned sh    = (ch == 0) ? 31u : (unsigned)(ch - 1);
        const unsigned smask = (rot << sh) & 0x80000000u;     // +-1 as sign flip
        acc += __uint_as_float(__float_as_uint(pix[ch]) ^ smask);
    }
    out[p] = acc;
}

extern "C" void kernel_launch(void* const* d_in, const int* in_sizes, int n_in,
                              void* d_out, int out_size, void* d_ws, size_t ws_size,
                              hipStream_t stream)
{
    (void)in_sizes; (void)n_in; (void)out_size; (void)d_ws; (void)ws_size;
    const float* x = (const float*)d_in[0];   // [512,64,64,31] f32
    const float* w = (const float*)d_in[1];   // [1024] f32
    float* out = (float*)d_out;               // [512,64,64] f32

    coded_shift_mask_sum<<<NPIX / TILE_PIX, 256, 0, stream>>>(x, w, out);
}